// CSTRModel_57509612093762
// MI455X (gfx1250) — compile-verified
//
#include <hip/hip_runtime.h>
#include <stdint.h>

// Fixed model constants from the reference.
#define K_A 0.5616f
#define K_B 0.3126f
#define K_C 48.43f
#define K_D 0.507f
#define K_E 55.0f
#define K_F 0.1538f
#define K_G 90.0f
#define K_H 0.16f

#define CHUNK 1024            // steps per LDS buffer (16 KB of float4)
#define ASYNC_PER_CHUNK 32    // CHUNK/32 lanes, one b128 per lane per instr

// ---------------------------------------------------------------------------
// Kernel 1 (massively parallel, bandwidth-bound): per-step u-dependent
// coefficients.  For each step t:
//   F4      = a*x0 + b*x1 + g_t
//   F4 - F5 = a*x0 + e_t*x1 + h_t
// with a, b state-independent constants and
//   g_t = (UA1*(F*u0+G) + F1*Cp*T1 - (UA1+F1*Cp)*C) / lam
//   s_t = UA2 / (1 + UA2/(2*Cp*u1)) / lam   (reciprocal folded here)
//   e_t = b - s_t*D,  h_t = g_t - s_t*(E - T200)
// Stored padded as float4 {g, e, h, 0} for aligned b128 traffic.
// ---------------------------------------------------------------------------
__global__ void cstr_coef_kernel(const float2* __restrict__ u,
                                 const float* __restrict__ params,
                                 float4* __restrict__ coef, int T) {
  const float UA2 = params[2], Cp = params[3], lam = params[4];
  const float F1 = params[6], F3 = params[8], T1 = params[9], T200 = params[10];

  const float UA1    = K_H * (F1 + F3);
  const float Kn     = UA1 + F1 * Cp;
  const float invLam = 1.0f / lam;
  const float b      = -Kn * K_A * invLam;
  const float kg1    = UA1 * K_F * invLam;
  const float kg0    = (UA1 * K_G + F1 * Cp * T1 - Kn * K_C) * invLam;
  const float twoCp  = 2.0f * Cp;
  const float c5     = UA2 * twoCp * invLam;   // s_t/lam = c5*u1 / (2Cp*u1 + UA2)
  const float emt    = K_E - T200;

  int t = blockIdx.x * blockDim.x + threadIdx.x;
  if (t < T) {
    float2 uv  = u[t];
    float  g   = fmaf(kg1, uv.x, kg0);
    float  den = fmaf(twoCp, uv.y, UA2);
    float  sl  = (c5 * uv.y) / den;
    float  e   = fmaf(-K_D, sl, b);
    float  h   = fmaf(-emt, sl, g);
    coef[t] = make_float4(g, e, h, 0.0f);
  }
}

// ---------------------------------------------------------------------------
// CDNA5 async copy: one b128 per lane, global -> LDS, tracked by ASYNCcnt.
// Issues exactly ASYNC_PER_CHUNK instructions per call (addresses clamped so
// the count is wave-uniform and s_wait_asynccnt thresholds are exact).
// ---------------------------------------------------------------------------
__device__ __forceinline__ void issue_chunk_async(const float4* __restrict__ coef,
                                                  float4* lbuf, int start, int T,
                                                  int lane) {
#pragma unroll
  for (int k = 0; k < ASYNC_PER_CHUNK; ++k) {
    int idx = start + k * 32 + lane;
    idx = idx < (T - 1) ? idx : (T - 1);  // clamp: always a valid address
    uint32_t lds_off = (uint32_t)(uintptr_t)(&lbuf[k * 32 + lane]);
    uint64_t gaddr   = (uint64_t)(uintptr_t)(coef + idx);
    asm volatile("global_load_async_to_lds_b128 %0, %1, off"
                 :: "v"(lds_off), "v"(gaddr)
                 : "memory");
  }
}

// ---------------------------------------------------------------------------
// Kernel 2: the irreducible serial scan.  One wave; lane 0 carries the 4-FMA
// dependence chain per step; all 32 lanes cooperate on the double-buffered
// async coefficient stream into LDS.
// ---------------------------------------------------------------------------
__global__ void __launch_bounds__(32, 1)
cstr_serial_kernel(const float4* __restrict__ coef,
                   const float* __restrict__ params,
                   const float* __restrict__ x0v,
                   float2* __restrict__ out, int T) {
  __shared__ float4 buf[2][CHUNK];  // 32 KB double buffer
  const int lane = threadIdx.x;

  const float M  = params[0], Cc = params[1], Cp = params[3], lam = params[4];
  const float F1 = params[6], X1 = params[7], F3 = params[8];

  const float UA1    = K_H * (F1 + F3);
  const float Kn     = UA1 + F1 * Cp;
  const float invLam = 1.0f / lam;
  const float a      = -Kn * K_B * invLam;
  const float b      = -Kn * K_A * invLam;
  const float invM   = 1.0f / M;
  const float invCc  = 1.0f / Cc;
  const float c1     = 1.0f - F1 * invM;   // x0' = x0*(c1 + invM*F4) + c2
  const float c2     = F1 * X1 * invM;

  float x0 = x0v[0];
  float x1 = x0v[1];

  const int nchunk = (T + CHUNK - 1) / CHUNK;
  issue_chunk_async(coef, buf[0], 0, T, lane);  // prefetch chunk 0

  float2* outp = out;
  for (int c = 0; c < nchunk; ++c) {
    const int pb = c & 1;
    if (c + 1 < nchunk) {
      // Overlap: stream chunk c+1 while we consume chunk c.
      issue_chunk_async(coef, buf[pb ^ 1], (c + 1) * CHUNK, T, lane);
      // In-order completion: <=32 outstanding  ==>  chunk c fully in LDS.
      asm volatile("s_wait_asynccnt 0x20" ::: "memory");
    } else {
      asm volatile("s_wait_asynccnt 0x0" ::: "memory");
    }

    const int base = c * CHUNK;
    const int n    = (T - base) < CHUNK ? (T - base) : CHUNK;
    if (lane == 0) {
#pragma unroll 4
      for (int i = 0; i < n; ++i) {
        float4 cf = buf[pb][i];                       // ds_load_b128
        float F4  = fmaf(a, x0, fmaf(b, x1, cf.x));   // A-side chain
        float m   = fmaf(a, x0, fmaf(cf.y, x1, cf.z));
        outp[i]   = make_float2(x0, x1);              // log pre-update state
        x0 = fmaf(x0, fmaf(invM, F4, c1), c2);
        x1 = fmaf(m, invCc, x1);
      }
    }
    outp += n;
  }
}

// ---------------------------------------------------------------------------
// Fallback (only if d_ws is too small): same serial math, coefficients
// computed inline from u.
// ---------------------------------------------------------------------------
__global__ void __launch_bounds__(32, 1)
cstr_serial_fallback(const float2* __restrict__ u,
                     const float* __restrict__ params,
                     const float* __restrict__ x0v,
                     float2* __restrict__ out, int T) {
  if (threadIdx.x != 0) return;
  const float M  = params[0], Cc = params[1], UA2 = params[2], Cp = params[3];
  const float lam = params[4], F1 = params[6], X1 = params[7], F3 = params[8];
  const float T1 = params[9], T200 = params[10];

  const float UA1    = K_H * (F1 + F3);
  const float Kn     = UA1 + F1 * Cp;
  const float invLam = 1.0f / lam;
  const float a      = -Kn * K_B * invLam;
  const float b      = -Kn * K_A * invLam;
  const float kg1    = UA1 * K_F * invLam;
  const float kg0    = (UA1 * K_G + F1 * Cp * T1 - Kn * K_C) * invLam;
  const float twoCp  = 2.0f * Cp;
  const float c5     = UA2 * twoCp * invLam;
  const float emt    = K_E - T200;
  const float invM   = 1.0f / M;
  const float invCc  = 1.0f / Cc;
  const float c1     = 1.0f - F1 * invM;
  const float c2     = F1 * X1 * invM;

  float x0 = x0v[0], x1 = x0v[1];
  for (int t = 0; t < T; ++t) {
    float2 uv  = u[t];
    float  g   = fmaf(kg1, uv.x, kg0);
    float  den = fmaf(twoCp, uv.y, UA2);
    float  sl  = (c5 * uv.y) / den;
    float  e   = fmaf(-K_D, sl, b);
    float  h   = fmaf(-emt, sl, g);
    float  F4  = fmaf(a, x0, fmaf(b, x1, g));
    float  m   = fmaf(a, x0, fmaf(e, x1, h));
    out[t] = make_float2(x0, x1);
    x0 = fmaf(x0, fmaf(invM, F4, c1), c2);
    x1 = fmaf(m, invCc, x1);
  }
}

extern "C" void kernel_launch(void* const* d_in, const int* in_sizes, int n_in,
                              void* d_out, int out_size, void* d_ws, size_t ws_size,
                              hipStream_t stream) {
  const float* u      = (const float*)d_in[0];   // [T,2] f32
  const float* x0     = (const float*)d_in[1];   // [2]   f32
  const float* params = (const float*)d_in[2];   // [11]  f32
  const int T = in_sizes[0] / 2;
  float2* out = (float2*)d_out;

  const size_t need = (size_t)T * sizeof(float4);
  if (d_ws != nullptr && ws_size >= need) {
    float4* coef = (float4*)d_ws;
    const int threads = 256;
    const int blocks  = (T + threads - 1) / threads;
    cstr_coef_kernel<<<blocks, threads, 0, stream>>>((const float2*)u, params, coef, T);
    cstr_serial_kernel<<<1, 32, 0, stream>>>(coef, params, x0, out, T);
  } else {
    cstr_serial_fallback<<<1, 32, 0, stream>>>((const float2*)u, params, x0, out, T);
  }
}